// RNN_88708254531777
// MI455X (gfx1250) — compile-verified
//
#include <hip/hip_runtime.h>

// LSTM fused kernel for MI455X (gfx1250, wave32).
// B=4096, T=200, F=64, H=16, O=1.
// One wave32 per 16-row batch tile; weights register-resident in WMMA B-layout;
// 80x v_wmma_f32_16x16x4_f32 per timestep; LDS used only to transpose h
// (D-layout -> A-layout) between timesteps.

typedef float v2f __attribute__((ext_vector_type(2)));
typedef float v8f __attribute__((ext_vector_type(8)));

#define T_STEPS 200
#define F_DIM   64
#define H_DIM   16
#define B_DIM   4096

__device__ __forceinline__ float fast_sigmoid(float x) {
    return __builtin_amdgcn_rcpf(1.0f + __expf(-x));
}
__device__ __forceinline__ float fast_tanh(float x) {
    // tanh(x) = 1 - 2/(exp(2x)+1); saturates correctly for |x| large.
    float e = __expf(2.0f * x);
    return 1.0f - 2.0f * __builtin_amdgcn_rcpf(e + 1.0f);
}

__global__ __launch_bounds__(32)
void lstm_fused_kernel(const float* __restrict__ X,     // [B,T,F]
                       const float* __restrict__ Wih,   // [4H,F]
                       const float* __restrict__ Whh,   // [4H,H]
                       const float* __restrict__ bih,   // [4H]
                       const float* __restrict__ bhh,   // [4H]
                       const float* __restrict__ Wout,  // [1,H]
                       const float* __restrict__ bout,  // [1]
                       float* __restrict__ out)         // [B,1]
{
    __shared__ __align__(16) float hbuf[16 * 16];  // h tile, [m][n]

    const int l    = threadIdx.x;      // 0..31
    const int lm   = l & 15;           // lane within half
    const int half = l >> 4;           // 0 or 1
    const int row0 = blockIdx.x * 16;  // batch tile base row

    // ---- load weights into registers (WMMA B-matrix layout chunks) ----
    // B chunk for k-base kb: vgpr v, lane l holds B[kb + 2*half + v][n=lm],
    // where B[k][n] = W[gate*16 + n][k]  (consecutive k => contiguous float2).
    v2f wih[4][16];
#pragma unroll
    for (int g = 0; g < 4; ++g)
#pragma unroll
        for (int kc = 0; kc < 16; ++kc)
            wih[g][kc] = *(const v2f*)&Wih[(g * 16 + lm) * F_DIM + kc * 4 + 2 * half];

    v2f whh[4][4];
#pragma unroll
    for (int g = 0; g < 4; ++g)
#pragma unroll
        for (int kc = 0; kc < 4; ++kc)
            whh[g][kc] = *(const v2f*)&Whh[(g * 16 + lm) * H_DIM + kc * 4 + 2 * half];

    float bias[4];
#pragma unroll
    for (int g = 0; g < 4; ++g)
        bias[g] = bih[g * 16 + lm] + bhh[g * 16 + lm];

    // per-lane x pointer: lane covers batch row (row0+lm), column parity 2*half
    const float* xlane = X + (size_t)(row0 + lm) * T_STEPS * F_DIM + 2 * half;

    // ---- state ----
    v2f hA[4];                 // h in A-matrix layout (K-chunks of 4)
#pragma unroll
    for (int kc = 0; kc < 4; ++kc) hA[kc] = (v2f){0.0f, 0.0f};
    float c[8];
#pragma unroll
    for (int r = 0; r < 8; ++r) c[r] = 0.0f;

    // ---- x double buffers (A-matrix layout chunks) ----
    v2f xa[16], xb[16];
#pragma unroll
    for (int kc = 0; kc < 16; ++kc)
        xa[kc] = *(const v2f*)&xlane[kc * 4];   // t = 0

    auto step = [&](v2f (&xcur)[16], v2f (&xnext)[16], int t) {
        // prefetch x for t+1 (clamped; issues loads early, overlaps WMMAs)
        const int tn = (t + 1 < T_STEPS) ? (t + 1) : (T_STEPS - 1);
        const float* xp = xlane + (size_t)tn * F_DIM;
#pragma unroll
        for (int kc = 0; kc < 16; ++kc)
            xnext[kc] = *(const v2f*)&xp[kc * 4];

        // init accumulators with (b_ih + b_hh) broadcast along rows
        v8f acc[4];
#pragma unroll
        for (int g = 0; g < 4; ++g) {
            v8f a;
#pragma unroll
            for (int r = 0; r < 8; ++r) a[r] = bias[g];
            acc[g] = a;
        }

        // gx contribution: Xtile[16x64] @ Wih^T[64x64] -> 4 gate tiles, K=64
#pragma unroll
        for (int kc = 0; kc < 16; ++kc)
#pragma unroll
            for (int g = 0; g < 4; ++g)
                acc[g] = __builtin_amdgcn_wmma_f32_16x16x4_f32(
                    false, xcur[kc], false, wih[g][kc], (short)0, acc[g], false, false);

        // recurrent contribution: h[16x16] @ Whh^T[16x64], K=16
#pragma unroll
        for (int kc = 0; kc < 4; ++kc)
#pragma unroll
            for (int g = 0; g < 4; ++g)
                acc[g] = __builtin_amdgcn_wmma_f32_16x16x4_f32(
                    false, hA[kc], false, whh[g][kc], (short)0, acc[g], false, false);

        // elementwise gate math: fully lane-local (i/f/g/o tiles share layout)
        float hD[8];
#pragma unroll
        for (int r = 0; r < 8; ++r) {
            float iv = fast_sigmoid(acc[0][r]);
            float fv = fast_sigmoid(acc[1][r]);
            float gv = fast_tanh(acc[2][r]);
            float ov = fast_sigmoid(acc[3][r]);
            c[r]  = fv * c[r] + iv * gv;
            hD[r] = ov * fast_tanh(c[r]);
        }

        // transpose h: D-layout -> A-layout via LDS
        __syncthreads();  // WAR vs previous iteration's hbuf loads
#pragma unroll
        for (int r = 0; r < 8; ++r)
            hbuf[(r + 8 * half) * 16 + lm] = hD[r];   // hbuf[M][N]
        __syncthreads();
#pragma unroll
        for (int kc = 0; kc < 4; ++kc)
            hA[kc] = *(const v2f*)&hbuf[lm * 16 + kc * 4 + 2 * half];
    };

#pragma unroll 1
    for (int t = 0; t < T_STEPS; t += 2) {
        step(xa, xb, t);
        step(xb, xa, t + 1);
    }

    // ---- final projection: out[b] = sigmoid(h_T[b,:] . Wout + bout) ----
    __syncthreads();  // hbuf holds h_T from the last step
    if (l < 16) {
        float s = bout[0];
#pragma unroll
        for (int n = 0; n < H_DIM; ++n)
            s += hbuf[l * 16 + n] * Wout[n];
        out[row0 + l] = fast_sigmoid(s);
    }
}

extern "C" void kernel_launch(void* const* d_in, const int* in_sizes, int n_in,
                              void* d_out, int out_size, void* d_ws, size_t ws_size,
                              hipStream_t stream) {
    const float* X    = (const float*)d_in[0];
    const float* Wih  = (const float*)d_in[1];
    const float* Whh  = (const float*)d_in[2];
    const float* bih  = (const float*)d_in[3];
    const float* bhh  = (const float*)d_in[4];
    const float* Wout = (const float*)d_in[5];
    const float* bout = (const float*)d_in[6];
    float* out = (float*)d_out;

    dim3 grid(B_DIM / 16);
    dim3 block(32);
    hipLaunchKernelGGL(lstm_fused_kernel, grid, block, 0, stream,
                       X, Wih, Whh, bih, bhh, Wout, bout, out);
}